// KVCacheManager_8864812499506
// MI455X (gfx1250) — compile-verified
//
#include <hip/hip_runtime.h>

// KV-cache decode update for MI455X (gfx1250).
// Pure streaming copy (1.07 GB traffic, ~46us floor @ 23.3 TB/s) + tiny scatter.
// Optimal path: b128 VMEM with non-temporal cache hints; no LDS/TDM hop needed.

typedef __attribute__((ext_vector_type(4))) float f4;

namespace {
constexpr int    L = 2, B = 8, H = 8, S = 4096, D = 128;
constexpr size_t CACHE_FLOATS = (size_t)L * B * H * S * D;   // 67,108,864 floats = 256 MB
constexpr size_t CACHE_F4     = CACHE_FLOATS / 4;            // 16,777,216 float4
constexpr size_t TOTAL_F4     = 2 * CACHE_F4;                // K-out + V-out halves
constexpr int    COPY_BLOCK    = 256;                        // 8 waves (wave32)
constexpr int    F4_PER_THREAD = 4;                          // 64 B per lane
constexpr unsigned COPY_GRID =
    (unsigned)(TOTAL_F4 / ((size_t)COPY_BLOCK * F4_PER_THREAD)); // 32768 blocks
}

// ---------------------------------------------------------------------------
// Bulk copy: out[0 .. N) = k_caches, out[N .. 2N) = v_caches.
// Each block streams a contiguous 16 KB chunk; 1024 f4 per block divides the
// K/V boundary exactly, so the source-pointer select is block-uniform.
// NT hints on both sides: single-use 512 MB stream, don't thrash the 192 MB L2.
// ---------------------------------------------------------------------------
__global__ __launch_bounds__(COPY_BLOCK) void kv_copy(
    const float* __restrict__ kc,
    const float* __restrict__ vc,
    float* __restrict__ out) {
  const f4* __restrict__ ks = (const f4*)kc;
  const f4* __restrict__ vs = (const f4*)vc;
  f4* __restrict__ o = (f4*)out;

  size_t base = (size_t)blockIdx.x * ((size_t)COPY_BLOCK * F4_PER_THREAD)
              + threadIdx.x;
  // Block-uniform K vs V select (blocks never straddle the boundary).
  const f4* __restrict__ src = (base < CACHE_F4) ? ks : (vs - CACHE_F4);

#pragma unroll
  for (int j = 0; j < F4_PER_THREAD; ++j) {
    size_t idx = base + (size_t)j * COPY_BLOCK;   // coalesced 4 KB per step
    f4 val = __builtin_nontemporal_load(&src[idx]);    // global_load_b128  th:NT
    __builtin_nontemporal_store(val, &o[idx]);         // global_store_b128 th:NT
  }
}

// ---------------------------------------------------------------------------
// Scatter the new token: one wave32 per (l,b,h); lane t writes float4 t of the
// 128-float head row at position_ids[b], for both K and V halves of out.
// Runs after kv_copy on the same stream, so it overwrites the stale rows.
// ---------------------------------------------------------------------------
__global__ __launch_bounds__(32) void kv_scatter(
    const float* __restrict__ nk,
    const float* __restrict__ nv,
    const int* __restrict__ pos,
    float* __restrict__ out) {
  int lbh = blockIdx.x;               // (l*B + b)*H + h   in [0, 128)
  int b   = (lbh / H) % B;
  int p   = pos[b];                   // token position in [0, S)

  size_t rowOut = ((size_t)lbh * S + (size_t)p) * D;  // dest row in cache
  size_t rowIn  = (size_t)lbh * D;                    // src row (q=1)

  const f4* __restrict__ k4 = (const f4*)(nk + rowIn);
  const f4* __restrict__ v4 = (const f4*)(nv + rowIn);
  f4* __restrict__ ok = (f4*)(out + rowOut);                 // K half
  f4* __restrict__ ov = (f4*)(out + CACHE_FLOATS + rowOut);  // V half

  int t = threadIdx.x;                // 0..31, covers D=128 as 32 float4
  ok[t] = k4[t];
  ov[t] = v4[t];
}

extern "C" void kernel_launch(void* const* d_in, const int* /*in_sizes*/, int /*n_in*/,
                              void* d_out, int /*out_size*/, void* /*d_ws*/,
                              size_t /*ws_size*/, hipStream_t stream) {
  const float* k_caches   = (const float*)d_in[0];
  const float* v_caches   = (const float*)d_in[1];
  const float* new_keys   = (const float*)d_in[2];
  const float* new_values = (const float*)d_in[3];
  const int*   position   = (const int*)d_in[4];
  float*       out        = (float*)d_out;

  kv_copy<<<COPY_GRID, COPY_BLOCK, 0, stream>>>(k_caches, v_caches, out);
  kv_scatter<<<L * B * H, 32, 0, stream>>>(new_keys, new_values, position, out);
}